// IRAM_73048803770537
// MI455X (gfx1250) — compile-verified
//
#include <hip/hip_runtime.h>
#include <hip/hip_bf16.h>

typedef __attribute__((ext_vector_type(2))) float v2f;
typedef __attribute__((ext_vector_type(8))) float v8f;
typedef int v4i __attribute__((vector_size(16)));  // matches builtin param type

#define WMMA_F32(a, b, c) \
  __builtin_amdgcn_wmma_f32_16x16x4_f32(false, (a), false, (b), (short)0, (c), false, false)

// Async global->LDS DMA (ASYNCcnt path). Builtin takes (int4* src, int4* dst, imm, imm).
#define ASYNC_B128(g, l) \
  __builtin_amdgcn_global_load_async_to_lds_b128((v4i*)(g), (v4i*)(l), 0, 0)
#define WAIT_ASYNC() asm volatile("s_wait_asynccnt 0x0" ::: "memory")

// ---------------------------------------------------------------------------
// conv3x3 SAME, 256->128 channels, full res 128x128. One block = (co, h) row.
// ---------------------------------------------------------------------------
__global__ __launch_bounds__(128) void conv_full_kernel(
    const float* __restrict__ x, const float* __restrict__ w,
    const float* __restrict__ b, float* __restrict__ out) {
  __shared__ float srow[3][130];
  const int wx = threadIdx.x;   // 0..127 (W)
  const int h  = blockIdx.x;    // 0..127
  const int co = blockIdx.y;    // 0..127
  float acc = 0.f;
  for (int ci = 0; ci < 256; ++ci) {
    __syncthreads();
#pragma unroll
    for (int r = 0; r < 3; ++r) {
      int hh = h + r - 1;
      float v = (hh >= 0 && hh < 128) ? x[(ci * 128 + hh) * 128 + wx] : 0.f;
      srow[r][wx + 1] = v;
    }
    if (wx < 3) srow[wx][0] = 0.f;
    else if (wx < 6) srow[wx - 3][129] = 0.f;
    __syncthreads();
    const float* wp = w + (co * 256 + ci) * 9;  // uniform -> scalar loads
#pragma unroll
    for (int r = 0; r < 3; ++r)
#pragma unroll
      for (int c = 0; c < 3; ++c)
        acc = fmaf(srow[r][wx + c], wp[r * 3 + c], acc);
  }
  acc += b[co];
  out[(co * 128 + h) * 128 + wx] = fmaxf(acc, 0.f);
}

// ---------------------------------------------------------------------------
// conv3x3 SAME on x[:, :, ::2, ::2] (64x64), 256->128. One block = (co, h).
// ---------------------------------------------------------------------------
__global__ __launch_bounds__(64) void conv_ds_kernel(
    const float* __restrict__ x, const float* __restrict__ w,
    const float* __restrict__ b, float* __restrict__ out) {
  __shared__ float srow[3][66];
  const int wx = threadIdx.x;   // 0..63
  const int h  = blockIdx.x;    // 0..63
  const int co = blockIdx.y;    // 0..127
  float acc = 0.f;
  for (int ci = 0; ci < 256; ++ci) {
    __syncthreads();
#pragma unroll
    for (int r = 0; r < 3; ++r) {
      int hh = h + r - 1;  // downsampled row index
      float v = (hh >= 0 && hh < 64) ? x[(ci * 128 + 2 * hh) * 128 + 2 * wx] : 0.f;
      srow[r][wx + 1] = v;
    }
    if (wx < 3) srow[wx][0] = 0.f;
    else if (wx < 6) srow[wx - 3][65] = 0.f;
    __syncthreads();
    const float* wp = w + (co * 256 + ci) * 9;
#pragma unroll
    for (int r = 0; r < 3; ++r)
#pragma unroll
      for (int c = 0; c < 3; ++c)
        acc = fmaf(srow[r][wx + c], wp[r * 3 + c], acc);
  }
  acc += b[co];
  out[co * 4096 + h * 64 + wx] = fmaxf(acc, 0.f);
}

// ---------------------------------------------------------------------------
// Flash attention: p123 = softmax(p1(16384x128) @ p2(128x4096) / 128) @ p3(4096x128)
// Block = 256 threads (8 waves). Each wave owns a 16-row M tile; block streams
// 32-column blocks of p2/p3 through LDS via async global->LDS DMA (ASYNCcnt).
// fp32 WMMA 16x16x4 for both GEMMs.
// ---------------------------------------------------------------------------
__global__ __launch_bounds__(256) void attn_kernel(
    const float* __restrict__ p1, const float* __restrict__ p2,
    const float* __restrict__ p3, float* __restrict__ p123) {
  __shared__ float sP2[128 * 32];   // [k][n] block of p2
  __shared__ float sP3[32 * 128];   // [r][c] block of p3
  __shared__ float sS[8][16 * 32];  // per-wave P-tile staging (C-layout -> A-layout)

  const int tid  = threadIdx.x;
  const int wave = tid >> 5;
  const int lane = tid & 31;
  const int row0 = (blockIdx.x * 8 + wave) * 16;
  const int am   = lane & 15;          // M (for A) or N (for B/C) index
  const int kb   = (lane >> 4) << 1;   // K sub-offset within 4-chunk
  const int mofs = (lane >> 4) << 3;   // C-layout row offset (0 or 8)

  // A tile: 16x128 of p1, K in 32 chunks of 4
  v2f a[32];
#pragma unroll
  for (int j = 0; j < 32; ++j) {
    const float* ap = p1 + (row0 + am) * 128 + 4 * j + kb;
    a[j].x = ap[0];
    a[j].y = ap[1];
  }

  v8f O[8];
#pragma unroll
  for (int g = 0; g < 8; ++g) O[g] = (v8f){};
  float mr[8], lr[8];
#pragma unroll
  for (int j = 0; j < 8; ++j) { mr[j] = -1e30f; lr[j] = 0.f; }

  for (int nb = 0; nb < 4096; nb += 32) {
    __syncthreads();
    if (nb + 32 < 4096) {
      __builtin_prefetch(p2 + (tid >> 5) * 4096 + nb + 32, 0, 1);  // global_prefetch_b8
    }
    // Async DMA: p2 block (128x32) and p3 block (32x128), 16 KB each, b128/lane.
#pragma unroll
    for (int t = 0; t < 4; ++t) {
      int i = (t * 256 + tid) * 4;  // 0..4092, step 4 floats (16B)
      ASYNC_B128(p2 + (i >> 5) * 4096 + nb + (i & 31), sP2 + i);
      ASYNC_B128(p3 + nb * 128 + i, sP3 + i);
    }
    WAIT_ASYNC();
    __syncthreads();

    // GEMM1: S (16x32) = A (16x128) @ p2blk (128x32), scaled by 1/TAU
    v8f S[2];
#pragma unroll
    for (int nt = 0; nt < 2; ++nt) {
      v8f c = (v8f){};
#pragma unroll
      for (int j = 0; j < 32; ++j) {
        v2f bv;
        bv.x = sP2[(4 * j + kb) * 32 + nt * 16 + am];
        bv.y = sP2[(4 * j + kb + 1) * 32 + nt * 16 + am];
        c = WMMA_F32(a[j], bv, c);
      }
      S[nt] = c * (1.0f / 128.0f);
    }

    // Online softmax (rows live in 16-lane halves -> xor-shuffle reductions)
    float corr[8];
#pragma unroll
    for (int j = 0; j < 8; ++j) {
      float tm = fmaxf(S[0][j], S[1][j]);
      tm = fmaxf(tm, __shfl_xor(tm, 1, 32));
      tm = fmaxf(tm, __shfl_xor(tm, 2, 32));
      tm = fmaxf(tm, __shfl_xor(tm, 4, 32));
      tm = fmaxf(tm, __shfl_xor(tm, 8, 32));
      float mnew = fmaxf(mr[j], tm);
      corr[j] = __expf(mr[j] - mnew);
      mr[j] = mnew;
      float p0 = __expf(S[0][j] - mnew);
      float p1v = __expf(S[1][j] - mnew);
      S[0][j] = p0;
      S[1][j] = p1v;
      float ps = p0 + p1v;
      ps += __shfl_xor(ps, 1, 32);
      ps += __shfl_xor(ps, 2, 32);
      ps += __shfl_xor(ps, 4, 32);
      ps += __shfl_xor(ps, 8, 32);
      lr[j] = lr[j] * corr[j] + ps;
    }
#pragma unroll
    for (int g = 0; g < 8; ++g)
#pragma unroll
      for (int j = 0; j < 8; ++j) O[g][j] *= corr[j];

    // Bounce P (C-layout) through per-wave LDS to re-read in A-layout
#pragma unroll
    for (int nt = 0; nt < 2; ++nt)
#pragma unroll
      for (int j = 0; j < 8; ++j)
        sS[wave][(j + mofs) * 32 + nt * 16 + am] = S[nt][j];
    asm volatile("s_wait_dscnt 0x0" ::: "memory");  // wave-local LDS RAW fence

    // GEMM2: O (16x128) += P (16x32) @ p3blk (32x128)
#pragma unroll
    for (int c8 = 0; c8 < 8; ++c8) {
      v2f a2;
      a2.x = sS[wave][am * 32 + 4 * c8 + kb];
      a2.y = sS[wave][am * 32 + 4 * c8 + kb + 1];
#pragma unroll
      for (int g = 0; g < 8; ++g) {
        v2f b2;
        b2.x = sP3[(4 * c8 + kb) * 128 + g * 16 + am];
        b2.y = sP3[(4 * c8 + kb + 1) * 128 + g * 16 + am];
        O[g] = WMMA_F32(a2, b2, O[g]);
      }
    }
  }

  // Normalize and store p123
#pragma unroll
  for (int g = 0; g < 8; ++g)
#pragma unroll
    for (int j = 0; j < 8; ++j) {
      int r = row0 + j + mofs;
      p123[r * 128 + g * 16 + am] = O[g][j] / lr[j];
    }
}

// ---------------------------------------------------------------------------
// p4 = relu(p123 @ w4^T + b4); out[0:N] = p4.flat + x.flat ; out[N:2N] = x
// (flat index f = r*256 + c equals the NCHW reshape index exactly)
// w4 tile (64 cols x 128 k, 32 KB) is DMA'd to LDS once and shared by 8 waves.
// ---------------------------------------------------------------------------
__global__ __launch_bounds__(256) void proj_kernel(
    const float* __restrict__ p123, const float* __restrict__ w4,
    const float* __restrict__ b4, const float* __restrict__ x,
    float* __restrict__ out) {
  __shared__ float sW[64 * 128];  // rows = output cols c0..c0+63, 128 K each
  const int tid  = threadIdx.x;
  const int wave = tid >> 5;
  const int lane = tid & 31;
  const int row0 = (blockIdx.x * 8 + wave) * 16;
  const int c0   = blockIdx.y * 64;
  const int am   = lane & 15;
  const int kb   = (lane >> 4) << 1;
  const int mofs = (lane >> 4) << 3;

  // Async DMA the w4 tile (contiguous 8192 floats starting at w4 + c0*128)
#pragma unroll
  for (int t = 0; t < 8; ++t) {
    int i = (t * 256 + tid) * 4;
    ASYNC_B128(w4 + c0 * 128 + i, sW + i);
  }

  v2f a[32];
#pragma unroll
  for (int j = 0; j < 32; ++j) {
    const float* ap = p123 + (row0 + am) * 128 + 4 * j + kb;
    a[j].x = ap[0];
    a[j].y = ap[1];
  }

  WAIT_ASYNC();
  __syncthreads();

#pragma unroll
  for (int nt = 0; nt < 4; ++nt) {
    const int cl = nt * 16 + am;   // local output column 0..63
    const int cg = c0 + cl;        // global output column
    v8f acc = (v8f){};
#pragma unroll
    for (int j = 0; j < 32; ++j) {
      v2f bv;
      bv.x = sW[cl * 128 + 4 * j + kb];      // contiguous pair -> ds_load_b64
      bv.y = sW[cl * 128 + 4 * j + kb + 1];
      acc = WMMA_F32(a[j], bv, acc);
    }
    const float bb = b4[cg];
#pragma unroll
    for (int j = 0; j < 8; ++j) {
      int r = row0 + j + mofs;
      int f = r * 256 + cg;
      float xv = x[f];
      out[f] = fmaxf(acc[j] + bb, 0.f) + xv;
      out[4194304 + f] = xv;  // second tuple element: x
    }
  }
}

// ---------------------------------------------------------------------------
extern "C" void kernel_launch(void* const* d_in, const int* in_sizes, int n_in,
                              void* d_out, int out_size, void* d_ws, size_t ws_size,
                              hipStream_t stream) {
  (void)in_sizes; (void)n_in; (void)out_size; (void)ws_size;
  const float* x  = (const float*)d_in[0];
  const float* w1 = (const float*)d_in[1];
  const float* b1 = (const float*)d_in[2];
  const float* w2 = (const float*)d_in[3];
  const float* b2 = (const float*)d_in[4];
  const float* w3 = (const float*)d_in[5];
  const float* b3 = (const float*)d_in[6];
  const float* w4 = (const float*)d_in[7];
  const float* b4 = (const float*)d_in[8];
  float* out = (float*)d_out;

  float* p1   = (float*)d_ws;              // 16384 x 128
  float* p2   = p1 + 16384 * 128;          // 128 x 4096
  float* p3   = p2 + 128 * 4096;           // 4096 x 128
  float* p123 = p3 + 4096 * 128;           // 16384 x 128   (20 MB total)

  conv_full_kernel<<<dim3(128, 128), 128, 0, stream>>>(x, w1, b1, p1);
  conv_ds_kernel<<<dim3(64, 128), 64, 0, stream>>>(x, w2, b2, p2);
  conv_ds_kernel<<<dim3(64, 128), 64, 0, stream>>>(x, w3, b3, p3);
  attn_kernel<<<dim3(128), 256, 0, stream>>>(p1, p2, p3, p123);
  proj_kernel<<<dim3(128, 4), 256, 0, stream>>>(p123, w4, b4, x, out);
}